// GCN_22290880266463
// MI455X (gfx1250) — compile-verified
//
#include <hip/hip_runtime.h>

// GCN on MI455X (gfx1250, wave32).
// GEMMs use V_WMMA_F32_16X16X4_F32 (exact fp32, matches reference numerics).
// Aggregation is edge-parallel with f32 global atomics; feature matrices are
// L2-resident (25.6MB << 192MB) so gather/atomic traffic stays on-chip.

typedef __attribute__((ext_vector_type(2))) float v2f;
typedef __attribute__((ext_vector_type(4))) float v4f;
typedef __attribute__((ext_vector_type(8))) float v8f;

#define DFEAT 128   // D_IN == D_HID == 128

// ---------------- degree / normalization ----------------

__global__ void deg_init_kernel(float* __restrict__ deg, int n) {
    int i = blockIdx.x * blockDim.x + threadIdx.x;
    if (i < n) deg[i] = 1.0f;               // self-loop contribution
}

__global__ void deg_count_kernel(const long long* __restrict__ dst,
                                 float* __restrict__ deg, int E) {
    int e = blockIdx.x * blockDim.x + threadIdx.x;
    if (e < E) atomicAdd(&deg[(int)dst[e]], 1.0f);
}

__global__ void deg_rsqrt_kernel(float* __restrict__ deg, int n) {
    int i = blockIdx.x * blockDim.x + threadIdx.x;
    if (i < n) deg[i] = rsqrtf(deg[i]);     // deg >= 1 always (self-loops)
}

// ---------------- fp32 WMMA GEMM:  Y[n,NO] = X[n,128] @ W[128,NO] (+bias) ----------------
// Block: 256 threads = 8 waves, 128 rows. Wave owns a fixed 16-column tile;
// all 32 K-step B fragments live in registers, reused across 8 row tiles.

template <int NO>
__global__ __launch_bounds__(256) void gemm_wmma_kernel(
    const float* __restrict__ X, const float* __restrict__ W,
    const float* __restrict__ bias, float* __restrict__ Y, int n_rows)
{
    constexpr int K  = DFEAT;
    constexpr int CT = NO / 16;     // column tiles (8 or 2)
    constexpr int RT = 8;           // row tiles per block (128 rows)

    const int wave = threadIdx.x >> 5;
    const int lane = threadIdx.x & 31;
    const int m    = lane & 15;     // row-in-tile / col-in-tile per WMMA layout
    const int ksel = lane >> 4;     // 0: K+{0,1}, 1: K+{2,3}
    const int ct   = wave % CT;     // fixed per wave given t += 8 stepping
    const int col  = ct * 16 + m;

    // B fragments: lane m holds W[kb][col], W[kb+1][col]  (kb = 4*kk + 2*ksel)
    v2f bfrag[K / 4];
#pragma unroll
    for (int kk = 0; kk < K / 4; ++kk) {
        const int kb = kk * 4 + 2 * ksel;
        v2f b;
        b.x = W[kb * NO + col];
        b.y = W[(kb + 1) * NO + col];
        bfrag[kk] = b;
    }
    float bv = 0.0f;
    if (bias) bv = bias[col];

    for (int t = wave; t < RT * CT; t += 8) {
        const int rt      = t / CT;
        const int rowbase = blockIdx.x * 128 + rt * 16;
        if (rowbase >= n_rows) break;            // wave-uniform

        int row = rowbase + m;
        if (row > n_rows - 1) row = n_rows - 1;  // clamp (EXEC stays all-1s)
        const float* xr = X + (size_t)row * K + 2 * ksel;

        v8f acc = {};
#pragma unroll
        for (int kk = 0; kk < K / 4; ++kk) {
            v2f a = *(const v2f*)(xr + kk * 4);  // 8B-aligned float2
            acc = __builtin_amdgcn_wmma_f32_16x16x4_f32(
                false, a, false, bfrag[kk], (short)0, acc, false, false);
        }

        // D layout: VGPR r -> row rowbase + r + 8*ksel, col = ct*16 + m
#pragma unroll
        for (int r = 0; r < 8; ++r) {
            const int orow = rowbase + r + 8 * ksel;
            if (orow < n_rows) Y[(size_t)orow * NO + col] = acc[r] + bv;
        }
    }
}

// ---------------- aggregation ----------------

// agg[i,:] = h[i,:] * dinv[i]^2   (self-loop term; also zero-initializes agg)
__global__ void self_init_kernel(const float* __restrict__ h,
                                 const float* __restrict__ dinv,
                                 float* __restrict__ agg, int n) {
    int idx = blockIdx.x * blockDim.x + threadIdx.x;
    int i = idx >> 5, c = idx & 31;
    if (i < n) {
        float w = dinv[i]; w *= w;
        v4f v = *(const v4f*)(h + (size_t)i * DFEAT + c * 4);
        v4f o; o.x = v.x * w; o.y = v.y * w; o.z = v.z * w; o.w = v.w * w;
        *(v4f*)(agg + (size_t)i * DFEAT + c * 4) = o;
    }
}

// One wave per edge; float4 coalesced gather (512B/wave) + 4 f32 atomics/lane.
__global__ void scatter_kernel(const long long* __restrict__ src,
                               const long long* __restrict__ dst,
                               const float* __restrict__ dinv,
                               const float* __restrict__ h,
                               float* __restrict__ agg, int E) {
    int gid = blockIdx.x * blockDim.x + threadIdx.x;
    int e = gid >> 5, lane = gid & 31;
    if (e < E) {
        const int s = (int)src[e];
        const int d = (int)dst[e];
        const float w = dinv[s] * dinv[d];
        v4f v = *(const v4f*)(h + (size_t)s * DFEAT + lane * 4);
        float* ap = agg + (size_t)d * DFEAT + lane * 4;
        atomicAdd(ap + 0, v.x * w);
        atomicAdd(ap + 1, v.y * w);
        atomicAdd(ap + 2, v.z * w);
        atomicAdd(ap + 3, v.w * w);
    }
}

// out[i,:] = relu(agg[i,:] + b[:])
__global__ void bias_relu_kernel(const float* __restrict__ agg,
                                 const float* __restrict__ b,
                                 float* __restrict__ out, int n) {
    int idx = blockIdx.x * blockDim.x + threadIdx.x;
    int i = idx >> 5, c = idx & 31;
    if (i < n) {
        v4f v = *(const v4f*)(agg + (size_t)i * DFEAT + c * 4);
        v4f bb = *(const v4f*)(b + c * 4);
        v4f o;
        o.x = fmaxf(v.x + bb.x, 0.0f);
        o.y = fmaxf(v.y + bb.y, 0.0f);
        o.z = fmaxf(v.z + bb.z, 0.0f);
        o.w = fmaxf(v.w + bb.w, 0.0f);
        *(v4f*)(out + (size_t)i * DFEAT + c * 4) = o;
    }
}

// ---------------- launch ----------------

extern "C" void kernel_launch(void* const* d_in, const int* in_sizes, int n_in,
                              void* d_out, int out_size, void* d_ws, size_t ws_size,
                              hipStream_t stream) {
    const float*     x    = (const float*)d_in[0];
    const long long* edge = (const long long*)d_in[1];   // int64 [2, E]
    const float*     W1   = (const float*)d_in[2];
    const float*     b1   = (const float*)d_in[3];
    const float*     W2   = (const float*)d_in[4];
    const float*     b2   = (const float*)d_in[5];
    const float*     Wc   = (const float*)d_in[6];
    const float*     bc   = (const float*)d_in[7];
    float*           out  = (float*)d_out;

    const int n = in_sizes[0] / DFEAT;
    const int E = in_sizes[1] / 2;
    const long long* src = edge;
    const long long* dst = edge + E;

    // workspace: dinv [n] | bufH [n*128] | bufA [n*128]
    char*  ws   = (char*)d_ws;
    size_t off  = ((size_t)n * 4 + 255) & ~(size_t)255;
    float* dinv = (float*)ws;
    float* bufH = (float*)(ws + off);
    float* bufA = (float*)(ws + off + (size_t)n * DFEAT * 4);

    const unsigned tpb     = 256;
    const unsigned gN      = (n + tpb - 1) / tpb;
    const unsigned gE      = (E + tpb - 1) / tpb;
    const unsigned gNF     = ((unsigned)((size_t)n * 32 + tpb - 1)) / tpb; // n*32 lanes of float4
    const unsigned gEdgeW  = ((unsigned)((size_t)E * 32 + tpb - 1)) / tpb; // wave per edge
    const unsigned gGemm   = (n + 127) / 128;

    // normalization
    deg_init_kernel<<<gN, tpb, 0, stream>>>(dinv, n);
    deg_count_kernel<<<gE, tpb, 0, stream>>>(dst, dinv, E);
    deg_rsqrt_kernel<<<gN, tpb, 0, stream>>>(dinv, n);

    // layer 1: h = x@W1 ; agg = D^-1/2 A D^-1/2 h ; relu(+b1)
    gemm_wmma_kernel<DFEAT><<<gGemm, tpb, 0, stream>>>(x, W1, nullptr, bufH, n);
    self_init_kernel<<<gNF, tpb, 0, stream>>>(bufH, dinv, bufA, n);
    scatter_kernel<<<gEdgeW, tpb, 0, stream>>>(src, dst, dinv, bufH, bufA, E);
    bias_relu_kernel<<<gNF, tpb, 0, stream>>>(bufA, b1, bufH, n);

    // layer 2
    gemm_wmma_kernel<DFEAT><<<gGemm, tpb, 0, stream>>>(bufH, W2, nullptr, bufA, n);
    self_init_kernel<<<gNF, tpb, 0, stream>>>(bufA, dinv, bufH, n);
    scatter_kernel<<<gEdgeW, tpb, 0, stream>>>(src, dst, dinv, bufA, bufH, E);
    bias_relu_kernel<<<gNF, tpb, 0, stream>>>(bufH, b2, bufA, n);

    // classifier head: out = h2@Wc + bc   (NO = 32)
    gemm_wmma_kernel<32><<<gGemm, tpb, 0, stream>>>(bufA, Wc, bc, out, n);
}